// RealNVP_3822520894002
// MI455X (gfx1250) — compile-verified
//
#include <hip/hip_runtime.h>

typedef __attribute__((ext_vector_type(2))) float v2f;
typedef __attribute__((ext_vector_type(8))) float v8f;

#define LAYERS 15
#define SLOPE  0.01f
#define TILES  4          // 16-point tiles per wave per layer pass (64 pts/wave)

// leaky_relu(v) = max(v, 0.01*v): exact for all finite v, lowers to
// v_mul + v_max_num_f32 (dual-issuable, no cmp/cndmask/carry traffic).
__device__ __forceinline__ float leaky(float v) {
    return fmaxf(v, SLOPE * v);
}

// Evaluate one MLP (1 -> 32 -> 16 -> 1) for TILES tiles of 16 points each.
// Middle layer is computed TRANSPOSED:  D = W2^T (16feat x 32) @ H1^T (32 x 16pts)
// so the 16->1 output layer reduces in-lane + one cross-half shuffle, and the
// b2 bias rides in the WMMA accumulator initialization for free.
// x1v[t]: input scalar of point (lane%16) in tile t (duplicated across halves).
// outv[t]: MLP output for that point (duplicated across halves).
__device__ __forceinline__ void mlp_eval(const float* __restrict__ W1,  // 32
                                         const float* __restrict__ b1,  // 32
                                         const float* __restrict__ W2,  // 32x16 row-major
                                         const float* __restrict__ b2,  // 16
                                         const float* __restrict__ W3,  // 16
                                         const float* __restrict__ b3,  // 1
                                         int lane,
                                         const float* x1v, float* outv)
{
    const int n    = lane & 15;         // point (B-matrix N) / feature (A-matrix M)
    const int hi   = lane >> 4;         // 0 = lanes 0-15, 1 = lanes 16-31
    const int off2 = hi << 1;           // K-offset of this lane half

    // av[]: W2^T in f32 16x4 A-matrix layout (M=feature n, K=j).
    // wv[]/bb[]: W1/b1 slice so h1 registers land in the 4x16 B-matrix layout.
    float wv[16], bb[16], av[16];
#pragma unroll
    for (int r = 0; r < 16; ++r) {
        const int j = 4 * (r >> 1) + (r & 1) + off2;  // K index for reg r
        wv[r] = W1[j];
        bb[r] = b1[j];
        av[r] = W2[j * 16 + n];          // W2^T[M=n][K=j]
    }
    // Per-lane-half slices of b2 / W3 (feature dim M = r + 8*hi in D).
    float b2f[8], w3f[8];
#pragma unroll
    for (int r = 0; r < 8; ++r) {
        b2f[r] = b2[r + 8 * hi];
        w3f[r] = W3[r + 8 * hi];
    }
    const float b3s = b3[0];

#pragma unroll
    for (int t = 0; t < TILES; ++t) {
        const float x1 = x1v[t];

        // Layer 1: h1 = leaky_relu(x1 * W1 + b1), in H1^T B-matrix layout.
        float h1[16];
#pragma unroll
        for (int r = 0; r < 16; ++r)
            h1[r] = leaky(fmaf(x1, wv[r], bb[r]));

        // Layer 2 (transposed): D[feat][pt] via 8x V_WMMA_F32_16X16X4_F32.
        // Accumulator starts at b2 (C-layout entry for VGPR r IS feature r+8*hi),
        // so the bias add costs zero instructions.
        v8f c;
#pragma unroll
        for (int r = 0; r < 8; ++r) c[r] = b2f[r];
#pragma unroll
        for (int k = 0; k < 8; ++k) {
            v2f a = { av[2 * k], av[2 * k + 1] };   // W2^T tile (A)
            v2f b = { h1[2 * k], h1[2 * k + 1] };   // H1^T tile (B)
            c = __builtin_amdgcn_wmma_f32_16x16x4_f32(false, a, false, b,
                                                      (short)0, c, false, false);
        }

        // Layer 3: out[pt] = sum_feat leaky(D) * W3 + b3.
        // Lane l holds features r+8*hi of point l%16 -> 8 in-lane FMAs,
        // then one cross-half add combines the two feature halves.
        float v = 0.0f;
#pragma unroll
        for (int r = 0; r < 8; ++r)
            v = fmaf(leaky(c[r]), w3f[r], v);
        v += __shfl_xor(v, 16);
        outv[t] = v + b3s;
    }
}

__global__ void __launch_bounds__(256)
realnvp_kernel(const float* __restrict__ x,
               const float* __restrict__ scale_w, const float* __restrict__ scale_b,
               const float* __restrict__ sW1, const float* __restrict__ sb1,
               const float* __restrict__ sW2, const float* __restrict__ sb2,
               const float* __restrict__ sW3, const float* __restrict__ sb3,
               const float* __restrict__ tW1, const float* __restrict__ tb1,
               const float* __restrict__ tW2, const float* __restrict__ tb2,
               const float* __restrict__ tW3, const float* __restrict__ tb3,
               float* __restrict__ out, int N)
{
    const int lane = threadIdx.x & 31;
    const long long wid  = (long long)blockIdx.x * (blockDim.x >> 5) + (threadIdx.x >> 5);
    const long long base = wid * (16 * TILES);

    // Flow state: z1, z2, log-det per owned point (duplicated across halves).
    float z1[TILES], z2[TILES], ld[TILES];
#pragma unroll
    for (int t = 0; t < TILES; ++t) {
        long long p = base + t * 16 + (lane & 15);
        if (p >= N) p = N - 1;                 // clamp: keep EXEC all-1s for WMMA
        const float2 xv = ((const float2*)x)[p];
        z1[t] = xv.y;                          // init z1 = x[:,1]
        z2[t] = xv.x;                          // init z2 = x[:,0]
        ld[t] = 0.0f;
    }

    for (int l = 0; l < LAYERS; ++l) {
        const float sw  = scale_w[l];
        const float sbv = scale_b[l];

        float x1v[TILES];
#pragma unroll
        for (int t = 0; t < TILES; ++t) x1v[t] = z2[t];

        float souts[TILES], touts[TILES];
        mlp_eval(sW1 + l * 32, sb1 + l * 32, sW2 + l * 512, sb2 + l * 16,
                 sW3 + l * 16, sb3 + l, lane, x1v, souts);
        mlp_eval(tW1 + l * 32, tb1 + l * 32, tW2 + l * 512, tb2 + l * 16,
                 tW3 + l * 16, tb3 + l, lane, x1v, touts);

#pragma unroll
        for (int t = 0; t < TILES; ++t) {
            const float ls = tanhf(souts[t]) * sw + sbv;
            const float x2 = z1[t];
            z1[t] = x1v[t];                        // new z1 = old z2
            z2[t] = expf(ls) * x2 + touts[t];      // new z2
            ld[t] += ls;
        }
    }

    // Store: lanes 0-15 write (z1,z2) pairs, lanes 16-31 write log-det.
#pragma unroll
    for (int t = 0; t < TILES; ++t) {
        const long long p = base + t * 16 + (lane & 15);
        if (p < N) {
            if (lane < 16) {
                float2 zz; zz.x = z1[t]; zz.y = z2[t];
                ((float2*)out)[p] = zz;
            } else {
                out[2 * (long long)N + p] = ld[t];
            }
        }
    }
}

extern "C" void kernel_launch(void* const* d_in, const int* in_sizes, int n_in,
                              void* d_out, int out_size, void* d_ws, size_t ws_size,
                              hipStream_t stream) {
    const float* x       = (const float*)d_in[0];
    const float* scale_w = (const float*)d_in[1];
    const float* scale_b = (const float*)d_in[2];
    const float* sW1     = (const float*)d_in[3];
    const float* sb1     = (const float*)d_in[4];
    const float* sW2     = (const float*)d_in[5];
    const float* sb2     = (const float*)d_in[6];
    const float* sW3     = (const float*)d_in[7];
    const float* sb3     = (const float*)d_in[8];
    const float* tW1     = (const float*)d_in[9];
    const float* tb1     = (const float*)d_in[10];
    const float* tW2     = (const float*)d_in[11];
    const float* tb2     = (const float*)d_in[12];
    const float* tW3     = (const float*)d_in[13];
    const float* tb3     = (const float*)d_in[14];
    float* out = (float*)d_out;

    const int N = in_sizes[0] / 2;                 // 4,194,304 points
    const int pts_per_block = (256 / 32) * 16 * TILES;  // 8 waves * 64 pts = 512
    const int grid = (N + pts_per_block - 1) / pts_per_block;

    realnvp_kernel<<<grid, 256, 0, stream>>>(
        x, scale_w, scale_b, sW1, sb1, sW2, sb2, sW3, sb3,
        tW1, tb1, tW2, tb2, tW3, tb3, out, N);
}